// HeuristicBimodalCSRPool_39737037423402
// MI455X (gfx1250) — compile-verified
//
#include <hip/hip_runtime.h>
#include <hip/hip_bf16.h>
#include <math.h>

#define WAVES_PER_BLOCK 8
#define D_FEAT 64  // row width of x_mod / x_pool (reference: D = 64)

// One wave32 per CSR segment:
//  1) strided scan of x_proj[:,0] over [start,end), per-lane (maxVal, minIdx)
//  2) wave32 xor-butterfly all-reduce (max value, tie -> smallest index)
//  3) 256B row copy x_mod[bestI] -> out via async global->LDS->global DMA
__global__ __launch_bounds__(256) void csr_argmax_pool_kernel(
    const float* __restrict__ x_mod,
    const float* __restrict__ x_proj,
    const int*   __restrict__ csr_idx,
    float*       __restrict__ out_pool,
    float*       __restrict__ out_seen,
    int n_groups, int n_proj)
{
    __shared__ char lds_stage[WAVES_PER_BLOCK * D_FEAT * sizeof(float)];

    const int lane = threadIdx.x & 31;
    const int wave = threadIdx.x >> 5;
    const int g = blockIdx.x * WAVES_PER_BLOCK + wave;
    if (g >= n_groups) return;

    const int start = csr_idx[g];
    const int end   = csr_idx[g + 1];

    // Per-lane scan: strictly-greater keeps the earliest index within a lane.
    float bestV = -INFINITY;
    int   bestI = 0x7FFFFFFF;
    for (int j = start + lane; j < end; j += 32) {
        float v = x_proj[(size_t)j * (size_t)n_proj];  // FEAT = 0 column
        if (v > bestV) { bestV = v; bestI = j; }
    }

    // wave32 butterfly all-reduce: max value, ties broken toward min index.
    #pragma unroll
    for (int off = 16; off > 0; off >>= 1) {
        float ov = __shfl_xor(bestV, off, 32);
        int   oi = __shfl_xor(bestI, off, 32);
        if (ov > bestV || (ov == bestV && oi < bestI)) { bestV = ov; bestI = oi; }
    }

    float* dst_row = out_pool + (size_t)g * D_FEAT;

    if (end > start) {  // wave-uniform branch: EXEC stays all-ones for async ops
        // LDS byte offset = low 32 bits of the flat shared-space address
        // also keeps lds_stage alive.
        unsigned lds_off = (unsigned)(size_t)&lds_stage[(wave * D_FEAT + lane * 2) * sizeof(float)];

        // 32 lanes x 8B = 256B row: global -> LDS (ASYNCcnt path, no VGPR data)
        unsigned long long src =
            (unsigned long long)(size_t)(x_mod + (size_t)bestI * D_FEAT + lane * 2);
        asm volatile("global_load_async_to_lds_b64 %0, %1, off"
                     :: "v"(lds_off), "v"(src) : "memory");
        asm volatile("s_wait_asynccnt 0" ::: "memory");

        // LDS -> global output row
        unsigned long long dst =
            (unsigned long long)(size_t)(dst_row + lane * 2);
        asm volatile("global_store_async_from_lds_b64 %0, %1, off"
                     :: "v"(dst), "v"(lds_off) : "memory");

        if (lane == 0) out_seen[g] = 1.0f;
        asm volatile("s_wait_asynccnt 0" ::: "memory");
    } else {
        float2 z; z.x = 0.0f; z.y = 0.0f;
        *reinterpret_cast<float2*>(dst_row + lane * 2) = z;
        if (lane == 0) out_seen[g] = 0.0f;
    }
}

extern "C" void kernel_launch(void* const* d_in, const int* in_sizes, int n_in,
                              void* d_out, int out_size, void* d_ws, size_t ws_size,
                              hipStream_t stream) {
    // setup_inputs() order: x_main (unused by reference), x_mod, x_proj, csr_idx
    const float* x_mod  = (const float*)d_in[1];
    const float* x_proj = (const float*)d_in[2];
    const int*   csr    = (const int*)d_in[3];

    const int n_groups = in_sizes[3] - 1;                 // csr has n_groups+1 entries
    const int n_elem   = in_sizes[1] / D_FEAT;            // x_mod is (E, 64)
    const int n_proj   = (n_elem > 0) ? in_sizes[2] / n_elem : 8;

    float* out_pool = (float*)d_out;                      // n_groups * 64 floats
    float* out_seen = out_pool + (size_t)n_groups * D_FEAT;  // then n_groups flags

    const int blocks = (n_groups + WAVES_PER_BLOCK - 1) / WAVES_PER_BLOCK;
    csr_argmax_pool_kernel<<<blocks, 256, 0, stream>>>(
        x_mod, x_proj, csr, out_pool, out_seen, n_groups, n_proj);
}